// LocationKernels_89438398972642
// MI455X (gfx1250) — compile-verified
//
#include <hip/hip_runtime.h>
#include <math.h>

typedef __attribute__((ext_vector_type(2))) float v2f;
typedef __attribute__((ext_vector_type(4))) float v4f;
typedef __attribute__((ext_vector_type(8))) float v8f;

#define COLS_PER_BLOCK 1024
#define ROWS_PER_BLOCK 32

// Computes row[j] = sum_k w[k] * N(j/ln; mean_k, 0.2) for a 1024-col tile,
// then broadcasts that tile across ROWS_PER_BLOCK output rows.
// Weighted 6-term reduction done with V_WMMA_F32_16X16X4_F32 (two K=4 steps).
__global__ __launch_bounds__(256) void locate_rows_kernel(
    const float* __restrict__ w, float* __restrict__ out,
    int ln, int B) {
  __shared__ float rowseg[COLS_PER_BLOCK];

  const int tid  = threadIdx.x;
  const int lane = tid & 31;
  const int wv   = tid >> 5;            // wave id 0..7 (wave32)
  const bool lo  = lane < 16;
  const int colBase = blockIdx.x * COLS_PER_BLOCK;

  // 6 scalar weights (hit in cache; cost irrelevant)
  const float w0 = w[0], w1 = w[1], w2 = w[2], w3 = w[3], w4 = w[4], w5 = w[5];

  const float inv_ln  = 1.0f / (float)ln;
  const float inv_std = 5.0f;                 // 1 / 0.2
  const float norm    = 1.9947114020071635f;  // 1 / (0.2 * sqrt(2*pi))

  // A-matrix (16x4 f32, ISA layout): lanes 0-15 hold K=0 (v0), K=1 (v1);
  // lanes 16-31 hold K=2 (v0), K=3 (v1). All 16 M-rows identical == weights.
  v2f a0, a1;
  a0.x = lo ? w0 : w2;   a0.y = lo ? w1 : w3;
  a1.x = lo ? w4 : 0.f;  a1.y = lo ? w5 : 0.f;   // K-slots 2,3 of step 2 are zero

  const int n = lane & 15;
  #pragma unroll
  for (int pass = 0; pass < 8; ++pass) {
    const int segCol = wv * 128 + pass * 16 + n;   // this lane's N column
    const float pos = (float)(colBase + segCol) * inv_ln;

    // B-matrix (4x16 f32): lanes 0-15 -> K=0 (v0), K=1 (v1);
    // lanes 16-31 -> K=2 (v0), K=3 (v1). pdf for this lane's two K means.
    const float m0 = lo ? 0.0f : 0.4f;
    const float m1 = lo ? 0.2f : 0.6f;
    float z0 = (pos - m0)   * inv_std;
    float z1 = (pos - m1)   * inv_std;
    float z4 = (pos - 0.8f) * inv_std;   // step-2 K=0 mean (lanes 16-31 x 0 in A)
    float z5 = (pos - 1.0f) * inv_std;   // step-2 K=1 mean
    v2f b0, b1;
    b0.x = __expf(-0.5f * z0 * z0) * norm;
    b0.y = __expf(-0.5f * z1 * z1) * norm;
    b1.x = __expf(-0.5f * z4 * z4) * norm;
    b1.y = __expf(-0.5f * z5 * z5) * norm;

    v8f c = {};
    c = __builtin_amdgcn_wmma_f32_16x16x4_f32(false, a0, false, b0,
                                              (short)0, c, false, false);
    c = __builtin_amdgcn_wmma_f32_16x16x4_f32(false, a1, false, b1,
                                              (short)0, c, false, false);
    // Every D row is identical; D vgpr0, lanes 0-15 = {M=0, N=lane}.
    if (lo) rowseg[segCol] = c[0];
  }
  __syncthreads();

  // Phase 2: broadcast the 4 KB tile down the rows with NT b128 stores.
  const int cv = tid;                       // float4 slot; 256*16B == 4KB tile
  v4f val = ((const v4f*)rowseg)[cv];

  const int row0 = blockIdx.y * ROWS_PER_BLOCK;
  const int rows = (B - row0 < ROWS_PER_BLOCK) ? (B - row0) : ROWS_PER_BLOCK;
  const long long rowStrideV = (long long)(ln >> 2);
  v4f* __restrict__ outv = (v4f*)out;
  long long p = (long long)row0 * rowStrideV + (long long)(colBase >> 2) + cv;

  if (colBase + cv * 4 + 3 < ln) {
    for (int r = 0; r < rows; ++r) {
      __builtin_nontemporal_store(val, &outv[p]);   // global_store_b128 TH=NT
      p += rowStrideV;
    }
  }
}

extern "C" void kernel_launch(void* const* d_in, const int* in_sizes, int n_in,
                              void* d_out, int out_size, void* d_ws, size_t ws_size,
                              hipStream_t stream) {
  const float* w = (const float*)d_in[0];
  // inp is never read; derive shape: |inp| = B*L, |out| = B*(L-2)
  const long long inpN = (long long)in_sizes[1];
  const long long outN = (long long)out_size;
  const long long B    = (inpN - outN) / 2;      // 4096
  const long long ln   = outN / B;               // 8192

  dim3 block(256);
  dim3 grid((unsigned)((ln + COLS_PER_BLOCK - 1) / COLS_PER_BLOCK),
            (unsigned)((B + ROWS_PER_BLOCK - 1) / ROWS_PER_BLOCK));
  locate_rows_kernel<<<grid, block, 0, stream>>>(w, (float*)d_out,
                                                 (int)ln, (int)B);
}